// Attn_89601607729485
// MI455X (gfx1250) — compile-verified
//
#include <hip/hip_runtime.h>
#include <hip/hip_bf16.h>

typedef float v2f __attribute__((ext_vector_type(2)));
typedef float v8f __attribute__((ext_vector_type(8)));
typedef float f4  __attribute__((ext_vector_type(4)));

#define T_DIM 1024
#define B_DIM 64
#define H_DIM 1024

// ---------------------------------------------------------------------------
// Kernel 1: v[b,h] = sum_o hidden[b,o] * W[o,h]
//   M = B_DIM (64), N = H_DIM (1024), K = H_DIM (1024)
//   One wave (32 threads) per 16x16 output tile; V_WMMA_F32_16X16X4_F32,
//   256 K-steps of 4. f32 WMMA keeps full reference precision.
//
//   16x4 f32 A layout : lanes 0-15 -> M=lane, VGPR0=K=k, VGPR1=K=k+1
//                       lanes 16-31 -> M=lane-16, VGPR0=K=k+2, VGPR1=K=k+3
//   4x16 f32 B layout : lanes 0-15 -> N=lane, VGPR0=K=k, VGPR1=K=k+1
//                       lanes 16-31 -> N=lane-16, VGPR0=K=k+2, VGPR1=K=k+3
//   16x16 f32 D layout: VGPR j -> row m0 + 8*half + j, col n0 + (lane&15)
// ---------------------------------------------------------------------------
__global__ void __launch_bounds__(32)
gemm_hidden_W(const float* __restrict__ hidden,
              const float* __restrict__ W,
              float* __restrict__ v)
{
    const int n0   = blockIdx.x * 16;           // column tile in H
    const int m0   = blockIdx.y * 16;           // row tile in B
    const int lane = threadIdx.x;               // 0..31
    const int half = lane >> 4;                 // 0: K pair {k,k+1}; 1: {k+2,k+3}
    const int l    = lane & 15;

    const float* __restrict__ arow = hidden + (size_t)(m0 + l) * H_DIM;

    v8f acc = {};
    #pragma unroll 4
    for (int k = 0; k < H_DIM; k += 4) {
        const int ka = k + half * 2;
        v2f a, bf;
        a.x  = arow[ka];
        a.y  = arow[ka + 1];
        bf.x = W[(size_t)ka       * H_DIM + n0 + l];
        bf.y = W[(size_t)(ka + 1) * H_DIM + n0 + l];
        acc = __builtin_amdgcn_wmma_f32_16x16x4_f32(
                  /*neg_a=*/false, a, /*neg_b=*/false, bf,
                  /*c_mod=*/(short)0, acc,
                  /*reuse_a=*/false, /*reuse_b=*/false);
    }

    const int mrow = m0 + half * 8;
    #pragma unroll
    for (int j = 0; j < 8; ++j)
        v[(size_t)(mrow + j) * H_DIM + n0 + l] = acc[j];
}

// ---------------------------------------------------------------------------
// Kernel 2 (bandwidth-dominant): energies[b,t] = dot(enc[t,b,:], v[b,:])
//   One 256-thread block per (t,b) row: each thread one float4 of the
//   1024-float row. enc is 256 MB streamed once -> non-temporal b128 loads;
//   v (256 KB) stays hot in L2.
// ---------------------------------------------------------------------------
__global__ void __launch_bounds__(256)
energy_kernel(const float* __restrict__ enc,
              const float* __restrict__ v,
              float* __restrict__ energies)
{
    const int blk = blockIdx.x;
    const int t   = blk >> 6;          // / B_DIM
    const int b   = blk & (B_DIM - 1);

    const f4* __restrict__ row = (const f4*)(enc + ((size_t)t * B_DIM + b) * H_DIM);
    const f4* __restrict__ vr  = (const f4*)(v + (size_t)b * H_DIM);

    const int i = threadIdx.x;         // 0..255 == H_DIM/4 float4s
    f4 e = __builtin_nontemporal_load(row + i);   // streaming: bypass caches
    f4 w = vr[i];                                  // L2-resident

    float s = e.x * w.x + e.y * w.y + e.z * w.z + e.w * w.w;

    // wave32 reduction
    #pragma unroll
    for (int off = 16; off; off >>= 1)
        s += __shfl_xor(s, off, 32);

    __shared__ float red[8];
    if ((threadIdx.x & 31) == 0) red[threadIdx.x >> 5] = s;
    __syncthreads();
    if (threadIdx.x == 0) {
        float tot = red[0];
        #pragma unroll
        for (int j = 1; j < 8; ++j) tot += red[j];
        energies[(size_t)b * T_DIM + t] = tot;
    }
}

// ---------------------------------------------------------------------------
// Kernel 3: per-batch softmax over t, including the (softmax-invariant but
// reference-faithful) bias constant c[b] = dot(hidden[b,:], bias).
// One block per b; 256 threads handle T=1024 (4 elements each).
// ---------------------------------------------------------------------------
__global__ void __launch_bounds__(256)
softmax_kernel(const float* __restrict__ energies,
               const float* __restrict__ hidden,
               const float* __restrict__ bias,
               float* __restrict__ out)
{
    __shared__ float smem[8];
    __shared__ float bcast;

    const int b   = blockIdx.x;
    const int tid = threadIdx.x;

    // ---- c[b] = dot(hidden[b,:], bias) ----
    const f4* __restrict__ hrow = (const f4*)(hidden + (size_t)b * H_DIM);
    const f4* __restrict__ brow = (const f4*)bias;
    f4 hv = hrow[tid];
    f4 bv = brow[tid];
    float c = hv.x * bv.x + hv.y * bv.y + hv.z * bv.z + hv.w * bv.w;
    #pragma unroll
    for (int off = 16; off; off >>= 1) c += __shfl_xor(c, off, 32);
    if ((tid & 31) == 0) smem[tid >> 5] = c;
    __syncthreads();
    if (tid == 0) {
        float s = smem[0];
        #pragma unroll
        for (int j = 1; j < 8; ++j) s += smem[j];
        bcast = s;
    }
    __syncthreads();
    c = bcast;
    __syncthreads();

    // ---- load energies row (+c), find max ----
    const float* __restrict__ erow = energies + (size_t)b * T_DIM;
    float e[4];
    float m = -3.402823466e38f;
    #pragma unroll
    for (int j = 0; j < 4; ++j) {
        e[j] = erow[tid + j * 256] + c;
        m = fmaxf(m, e[j]);
    }
    #pragma unroll
    for (int off = 16; off; off >>= 1) m = fmaxf(m, __shfl_xor(m, off, 32));
    if ((tid & 31) == 0) smem[tid >> 5] = m;
    __syncthreads();
    if (tid == 0) {
        float s = smem[0];
        #pragma unroll
        for (int j = 1; j < 8; ++j) s = fmaxf(s, smem[j]);
        bcast = s;
    }
    __syncthreads();
    m = bcast;
    __syncthreads();

    // ---- exp + sum ----
    float sum = 0.f;
    #pragma unroll
    for (int j = 0; j < 4; ++j) {
        e[j] = __expf(e[j] - m);
        sum += e[j];
    }
    #pragma unroll
    for (int off = 16; off; off >>= 1) sum += __shfl_xor(sum, off, 32);
    if ((tid & 31) == 0) smem[tid >> 5] = sum;
    __syncthreads();
    if (tid == 0) {
        float s = smem[0];
        #pragma unroll
        for (int j = 1; j < 8; ++j) s += smem[j];
        bcast = s;
    }
    __syncthreads();
    const float inv = 1.f / bcast;

    // ---- write [B,1,T] ----
    float* __restrict__ orow = out + (size_t)b * T_DIM;
    #pragma unroll
    for (int j = 0; j < 4; ++j)
        orow[tid + j * 256] = e[j] * inv;
}

// ---------------------------------------------------------------------------
extern "C" void kernel_launch(void* const* d_in, const int* in_sizes, int n_in,
                              void* d_out, int out_size, void* d_ws, size_t ws_size,
                              hipStream_t stream)
{
    const float* hidden = (const float*)d_in[0];   // [1,B,H]
    const float* enc    = (const float*)d_in[1];   // [T,B,H]
    const float* W      = (const float*)d_in[2];   // [H,H]
    const float* bias   = (const float*)d_in[3];   // [H]
    float* out = (float*)d_out;                    // [B,1,T]

    float* v        = (float*)d_ws;                // B*H floats   (256 KB)
    float* energies = v + (size_t)B_DIM * H_DIM;   // B*T floats   (256 KB)

    // 1) v = hidden @ W  (WMMA f32 GEMM, 4x64 tiles of 16x16, one wave each)
    gemm_hidden_W<<<dim3(H_DIM / 16, B_DIM / 16), 32, 0, stream>>>(hidden, W, v);

    // 2) energies[b,t] = dot(enc[t,b,:], v[b,:])  -- 256 MB stream, NT loads
    energy_kernel<<<T_DIM * B_DIM, 256, 0, stream>>>(enc, v, energies);

    // 3) softmax over t (+ bias constant), write [B,1,T]
    softmax_kernel<<<B_DIM, 256, 0, stream>>>(energies, hidden, bias, out);
}